// Net_cora_34857954574878
// MI455X (gfx1250) — compile-verified
//
#include <hip/hip_runtime.h>
#include <hip/hip_bf16.h>

// Problem constants (from reference)
#define NN 1200          // nodes
#define EE 19200         // edges
#define HH 20            // hidden dim
#define FF 1433          // feature dim
#define QQ 2048          // query pairs
#define WPR ((NN + 31) / 32)   // adjacency bitset words per row

typedef __attribute__((ext_vector_type(2))) float vf2;
typedef __attribute__((ext_vector_type(8))) float vf8;

// ---------------------------------------------------------------------------
// Graph preprocessing (all deterministic: int atomics + stable counting sort)
// ---------------------------------------------------------------------------
__global__ __launch_bounds__(256) void edge_count_kernel(
    const int* __restrict__ ei, int* __restrict__ cnt_src,
    int* __restrict__ cnt_dst, unsigned* __restrict__ adjbits) {
  int e = blockIdx.x * blockDim.x + threadIdx.x;
  if (e >= EE) return;
  int s = ei[e], d = ei[EE + e];
  atomicAdd(&cnt_src[s], 1);
  atomicAdd(&cnt_dst[d], 1);
  atomicOr(&adjbits[s * WPR + (d >> 5)], 1u << (d & 31));
}

__global__ __launch_bounds__(256) void dinv_kernel(
    const int* __restrict__ cnt_dst, float* __restrict__ dinv) {
  int i = blockIdx.x * blockDim.x + threadIdx.x;
  if (i >= NN) return;
  dinv[i] = rsqrtf((float)cnt_dst[i] + 1.0f);   // deg = in-count + 1
}

__global__ __launch_bounds__(64) void scan_kernel(
    const int* __restrict__ cs, const int* __restrict__ cd,
    int* __restrict__ os, int* __restrict__ od) {
  if (threadIdx.x == 0) { int a = 0; for (int i = 0; i < NN; i++) { os[i] = a; a += cs[i]; } os[NN] = a; }
  if (threadIdx.x == 1) { int a = 0; for (int i = 0; i < NN; i++) { od[i] = a; a += cd[i]; } od[NN] = a; }
}

// Stable rank among equal keys -> deterministic CSR edge ordering
__global__ __launch_bounds__(128) void rank_place_kernel(
    const int* __restrict__ ei, const int* __restrict__ os,
    const int* __restrict__ od, int* __restrict__ srt_s, int* __restrict__ srt_d) {
  int e = blockIdx.x * blockDim.x + threadIdx.x;
  if (e >= EE) return;
  int s = ei[e], d = ei[EE + e];
  int rs = 0, rd = 0;
  for (int e2 = 0; e2 < e; ++e2) {
    rs += (ei[e2] == s);
    rd += (ei[EE + e2] == d);
  }
  srt_s[os[s] + rs] = e;
  srt_d[od[d] + rd] = e;
}

// Zero-pad B (K x Nc, ldb == Nc) into Bp (Kp x 32), Kp = ceil4(K).
// Makes all B accesses in the GEMM K-loop unconditional, and makes the
// K-tail safe (0 * finite == 0).
__global__ __launch_bounds__(256) void pad_b_kernel(
    const float* __restrict__ B, float* __restrict__ Bp, int K, int Nc, int Kp) {
  int idx = blockIdx.x * blockDim.x + threadIdx.x;
  if (idx >= Kp * 32) return;
  int k = idx >> 5, n = idx & 31;
  Bp[idx] = (k < K && n < Nc) ? B[k * Nc + n] : 0.0f;
}

// ---------------------------------------------------------------------------
// f32 WMMA GEMM: D[M,N] = A[M,K] @ B[K,N] (+ bias[N]), row-major.
// REQUIRES: M % 16 == 0 (true for all call sites: 1200, 19200).
// B comes pre-padded to Kp x 32 with zeros (see pad_b_kernel).
// One wave32 per 16x16 tile, K stepped by 4 via V_WMMA_F32_16X16X4_F32.
// A frag (ISA 32-bit 16x4): lanes 0-15 -> K+0/1, lanes 16-31 -> K+2/3.
// C/D frag: VGPR v, lane l -> m = v + 8*(l/16), n = l%16.
// Inner loop is branch-free: unconditional loads, pointer increments only.
// ---------------------------------------------------------------------------
__global__ __launch_bounds__(32) void gemm_wmma_f32(
    const float* __restrict__ A, const float* __restrict__ Bp,
    const float* __restrict__ bias, float* __restrict__ D,
    int M, int K, int Nc, int lda, int ldd) {
  int tm = blockIdx.x * 16;
  int tn = blockIdx.y * 16;
  int lane = threadIdx.x;
  int half = lane >> 4;        // 0: K+0/1, 1: K+2/3
  int l16  = lane & 15;
  int m = tm + l16;            // always < M
  int n = tn + l16;            // < 32; D-store guarded by n < Nc
  int arow = m * lda;          // fits in int for all call sites
  const float* ap = A + arow + half * 2;
  const float* bp = Bp + half * 2 * 32 + n;
  vf8 acc = {};
  int K4 = K & ~3;
  for (int k0 = 0; k0 < K4; k0 += 4) {
    vf2 a, b;
    a.x = ap[0];
    a.y = ap[1];
    b.x = bp[0];
    b.y = bp[32];
    ap += 4;
    bp += 128;
    acc = __builtin_amdgcn_wmma_f32_16x16x4_f32(
        false, a, false, b, (short)0, acc, false, false);
  }
  if (K4 < K) {                // K-tail (only feat GEMM, K=1433)
    int ka = K4 + half * 2;
    int c0 = ka     < K ? ka     : K - 1;   // clamped, in-bounds addresses
    int c1 = ka + 1 < K ? ka + 1 : K - 1;
    float v0 = A[arow + c0];
    float v1 = A[arow + c1];
    vf2 a, b;
    a.x = (ka     < K) ? v0 : 0.0f;         // v_cndmask, no branch
    a.y = (ka + 1 < K) ? v1 : 0.0f;
    b.x = bp[0];                            // padded zeros beyond K
    b.y = bp[32];
    acc = __builtin_amdgcn_wmma_f32_16x16x4_f32(
        false, a, false, b, (short)0, acc, false, false);
  }
  if (n < Nc) {
    float bv = bias ? bias[n] : 0.0f;
#pragma unroll
    for (int v = 0; v < 8; ++v) {
      int mr = tm + v + 8 * half;
      D[mr * ldd + n] = acc[v] + bv;
    }
  }
}

// ---------------------------------------------------------------------------
// GCN aggregation: hout[i,:] = bg + t[i,:]*dinv[i]^2 +
//                  sum_{e: dst(e)=i} t[src(e),:]*dinv[src]*dinv[i]
// One wave per node; lane = feature dim; CSR order -> deterministic.
// ---------------------------------------------------------------------------
__global__ __launch_bounds__(32) void aggregate_kernel(
    const float* __restrict__ t, const float* __restrict__ dinv,
    const int* __restrict__ off_dst, const int* __restrict__ srt_dst,
    const int* __restrict__ ei, const float* __restrict__ bg,
    float* __restrict__ hout) {
  int i = blockIdx.x;
  int d = threadIdx.x;
  if (d >= HH) return;
  float di  = dinv[i];
  float acc = t[i * HH + d] * di * di + bg[d];
  int p0 = off_dst[i], p1 = off_dst[i + 1];
  for (int p = p0; p < p1; ++p) {
    int e = srt_dst[p];
    int s = ei[e];
    acc += t[s * HH + d] * dinv[s] * di;
  }
  hout[i * HH + d] = acc;
}

// val[e] = concat(h2[src(e)], h2[dst(e)])
__global__ __launch_bounds__(256) void build_val_kernel(
    const int* __restrict__ ei, const float* __restrict__ h2,
    float* __restrict__ val) {
  int idx = blockIdx.x * blockDim.x + threadIdx.x;
  if (idx >= EE * 2 * HH) return;
  int e = idx / (2 * HH);
  int c = idx - e * (2 * HH);
  int node = (c < HH) ? ei[e] : ei[EE + e];
  int cc   = (c < HH) ? c : (c - HH);
  val[idx] = h2[node * HH + cc];
}

// ---------------------------------------------------------------------------
// Per directed query pair (a,b):
//   C[a,b,h] = sum_{e1 in out(a)} sum_{e2 in in(b), src(e2)=dst(e1)} A1*A2
//   suppC    = exists k in out(a) with adj[k,b]
//   Y        = (suppC | adj) ? [C, adj] @ W3 + b3 : 0
// One wave per pair; lane = output dim h; LDS broadcast for the 21-vector.
// ---------------------------------------------------------------------------
__global__ __launch_bounds__(32) void pair_kernel(
    const int* __restrict__ pos1, const int* __restrict__ pos2,
    const int* __restrict__ ei,
    const int* __restrict__ off_src, const int* __restrict__ srt_src,
    const int* __restrict__ off_dst, const int* __restrict__ srt_dst,
    const unsigned* __restrict__ adjbits,
    const float* __restrict__ A1, const float* __restrict__ A2,
    const float* __restrict__ W3, const float* __restrict__ b3,
    float* __restrict__ Y) {
  int p = blockIdx.x;            // directed pair index, 0..2*QQ-1
  int q = p >> 1;
  int i0 = pos1[pos2[2 * q]];
  int j0 = pos1[pos2[2 * q + 1]];
  int a = (p & 1) ? j0 : i0;
  int b = (p & 1) ? i0 : j0;
  int lane = threadIdx.x;

  bool adj_ab = (adjbits[a * WPR + (b >> 5)] >> (b & 31)) & 1u;

  int o0 = off_src[a], o1 = off_src[a + 1];
  int d0 = off_dst[b], d1 = off_dst[b + 1];

  bool suppC = false;
  for (int pp = o0; pp < o1; ++pp) {
    int k = ei[EE + srt_src[pp]];                       // dst of out-edge of a
    if ((adjbits[k * WPR + (b >> 5)] >> (b & 31)) & 1u) suppC = true;
  }

  float c = 0.0f;
  for (int p1 = o0; p1 < o1; ++p1) {
    int e1 = srt_src[p1];
    int k  = ei[EE + e1];
    float a1 = (lane < HH) ? A1[e1 * HH + lane] : 0.0f;
    for (int p2 = d0; p2 < d1; ++p2) {
      int e2 = srt_dst[p2];
      if (ei[e2] == k) {
        float a2 = (lane < HH) ? A2[e2 * HH + lane] : 0.0f;
        c += a1 * a2;
      }
    }
  }

  __shared__ float sv[HH + 1];
  if (lane < HH) sv[lane] = c;
  if (lane == 0) sv[HH] = adj_ab ? 1.0f : 0.0f;
  __syncthreads();

  if (lane < HH) {
    bool supp = suppC || adj_ab;
    float y = 0.0f;
    if (supp) {
      y = b3[lane];
      for (int c2 = 0; c2 <= HH; ++c2) y += sv[c2] * W3[c2 * HH + lane];
    }
    Y[p * HH + lane] = y;
  }
}

// out[q] = bd + sum_h Y(i,j)[h]*Y(j,i)[h]*Wd[h] + sum_h h2[i][h]*h2[j][h]*Wd[HH+h]
__global__ __launch_bounds__(256) void out_kernel(
    const int* __restrict__ pos1, const int* __restrict__ pos2,
    const float* __restrict__ Y, const float* __restrict__ h2,
    const float* __restrict__ Wd, const float* __restrict__ bd,
    float* __restrict__ out) {
  int q = blockIdx.x * blockDim.x + threadIdx.x;
  if (q >= QQ) return;
  int i = pos1[pos2[2 * q]];
  int j = pos1[pos2[2 * q + 1]];
  float s = bd[0];
  for (int h = 0; h < HH; ++h)
    s += Y[(2 * q) * HH + h] * Y[(2 * q + 1) * HH + h] * Wd[h];
  for (int h = 0; h < HH; ++h)
    s += h2[i * HH + h] * h2[j * HH + h] * Wd[HH + h];
  out[q] = s;
}

// ---------------------------------------------------------------------------
extern "C" void kernel_launch(void* const* d_in, const int* in_sizes, int n_in,
                              void* d_out, int out_size, void* d_ws, size_t ws_size,
                              hipStream_t stream) {
  const int*   ei   = (const int*)d_in[1];
  const int*   pos1 = (const int*)d_in[2];
  const int*   pos2 = (const int*)d_in[3];
  const float* feat = (const float*)d_in[4];
  const float* Wg1  = (const float*)d_in[5];
  const float* bg1  = (const float*)d_in[6];
  const float* Wg2  = (const float*)d_in[7];
  const float* bg2  = (const float*)d_in[8];
  const float* W1   = (const float*)d_in[9];
  const float* b1   = (const float*)d_in[10];
  const float* W2   = (const float*)d_in[11];
  const float* b2   = (const float*)d_in[12];
  const float* W3   = (const float*)d_in[13];
  const float* b3   = (const float*)d_in[14];
  const float* Wd   = (const float*)d_in[15];
  const float* bd   = (const float*)d_in[16];
  float* out = (float*)d_out;

  char* ws = (char*)d_ws;
  size_t off = 0;
  auto alloc = [&](size_t bytes) {
    size_t o = off;
    off += (bytes + 255) & ~(size_t)255;
    return o;
  };

  const int KpF = (FF + 3) & ~3;   // 1436

  // cnt_src | cnt_dst | adjbits packed so one memset zeroes all three
  size_t zero_bytes = (size_t)(2 * NN + NN * WPR) * sizeof(int);
  int*      cnt_src = (int*)(ws + alloc(zero_bytes));
  int*      cnt_dst = cnt_src + NN;
  unsigned* adjbits = (unsigned*)(cnt_dst + NN);
  int* off_src = (int*)(ws + alloc((NN + 1) * sizeof(int)));
  int* off_dst = (int*)(ws + alloc((NN + 1) * sizeof(int)));
  int* srt_src = (int*)(ws + alloc((size_t)EE * sizeof(int)));
  int* srt_dst = (int*)(ws + alloc((size_t)EE * sizeof(int)));
  float* dinv  = (float*)(ws + alloc(NN * sizeof(float)));
  float* t     = (float*)(ws + alloc((size_t)NN * HH * sizeof(float)));
  float* h1    = (float*)(ws + alloc((size_t)NN * HH * sizeof(float)));
  float* h2    = (float*)(ws + alloc((size_t)NN * HH * sizeof(float)));
  float* val   = (float*)(ws + alloc((size_t)EE * 2 * HH * sizeof(float)));
  float* A1    = (float*)(ws + alloc((size_t)EE * HH * sizeof(float)));
  float* A2    = (float*)(ws + alloc((size_t)EE * HH * sizeof(float)));
  float* Y     = (float*)(ws + alloc((size_t)2 * QQ * HH * sizeof(float)));
  float* BpG1  = (float*)(ws + alloc((size_t)KpF * 32 * sizeof(float)));
  float* BpG2  = (float*)(ws + alloc((size_t)HH * 32 * sizeof(float)));
  float* BpW1  = (float*)(ws + alloc((size_t)2 * HH * 32 * sizeof(float)));
  float* BpW2  = (float*)(ws + alloc((size_t)2 * HH * 32 * sizeof(float)));

  hipMemsetAsync(cnt_src, 0, zero_bytes, stream);
  edge_count_kernel<<<(EE + 255) / 256, 256, 0, stream>>>(ei, cnt_src, cnt_dst, adjbits);
  dinv_kernel<<<(NN + 255) / 256, 256, 0, stream>>>(cnt_dst, dinv);
  scan_kernel<<<1, 64, 0, stream>>>(cnt_src, cnt_dst, off_src, off_dst);
  rank_place_kernel<<<(EE + 127) / 128, 128, 0, stream>>>(ei, off_src, off_dst, srt_src, srt_dst);

  // Pad all GEMM B-matrices to Kp x 32 with zeros
  pad_b_kernel<<<(KpF * 32 + 255) / 256, 256, 0, stream>>>(Wg1, BpG1, FF, HH, KpF);
  pad_b_kernel<<<(HH * 32 + 255) / 256, 256, 0, stream>>>(Wg2, BpG2, HH, HH, HH);
  pad_b_kernel<<<(2 * HH * 32 + 255) / 256, 256, 0, stream>>>(W1, BpW1, 2 * HH, HH, 2 * HH);
  pad_b_kernel<<<(2 * HH * 32 + 255) / 256, 256, 0, stream>>>(W2, BpW2, 2 * HH, HH, 2 * HH);

  dim3 b32(32);
  dim3 gN(NN / 16, 2);   // 75 x 2 tiles
  // GCN layer 1: t = feat @ Wg1, then aggregate
  gemm_wmma_f32<<<gN, b32, 0, stream>>>(feat, BpG1, nullptr, t, NN, FF, HH, FF, HH);
  aggregate_kernel<<<NN, 32, 0, stream>>>(t, dinv, off_dst, srt_dst, ei, bg1, h1);
  // GCN layer 2: t = h1 @ Wg2, then aggregate
  gemm_wmma_f32<<<gN, b32, 0, stream>>>(h1, BpG2, nullptr, t, NN, HH, HH, HH, HH);
  aggregate_kernel<<<NN, 32, 0, stream>>>(t, dinv, off_dst, srt_dst, ei, bg2, h2);

  // Edge features: val = [h2[src], h2[dst]]; A1 = val@W1+b1; A2 = val@W2+b2
  build_val_kernel<<<(EE * 2 * HH + 255) / 256, 256, 0, stream>>>(ei, h2, val);
  dim3 gE(EE / 16, 2);   // 1200 x 2 tiles
  gemm_wmma_f32<<<gE, b32, 0, stream>>>(val, BpW1, b1, A1, EE, 2 * HH, HH, 2 * HH, HH);
  gemm_wmma_f32<<<gE, b32, 0, stream>>>(val, BpW2, b2, A2, EE, 2 * HH, HH, 2 * HH, HH);

  // Sparse einsum only at the 4096 directed query pairs, then final dot
  pair_kernel<<<2 * QQ, 32, 0, stream>>>(pos1, pos2, ei, off_src, srt_src,
                                         off_dst, srt_dst, adjbits, A1, A2, W3, b3, Y);
  out_kernel<<<(QQ + 255) / 256, 256, 0, stream>>>(pos1, pos2, Y, h2, Wd, bd, out);
}